// DGCNNFeature_90374701842604
// MI455X (gfx1250) — compile-verified
//
#include <hip/hip_runtime.h>
#include <hip/hip_bf16.h>

#define N_PTS 16384
#define K_NB  16
#define H_DIM 64
#define NSEG  4                      // j-range split for the KNN scan
#define SEG_LEN (N_PTS / NSEG)       // 4096

typedef float v2f __attribute__((ext_vector_type(2)));
typedef float v8f __attribute__((ext_vector_type(8)));

// D = A(16x4 f32) * B(4x16 f32) + C(16x16 f32), full-precision WMMA.
// (neg_a, A, neg_b, B, c_mod, C, reuse_a, reuse_b) -- codegen-verified round 1.
static __device__ __forceinline__ v8f wmma_f32_16x16x4(v2f a, v2f b, v8f c) {
  return __builtin_amdgcn_wmma_f32_16x16x4_f32(false, a, false, b, (short)0, c,
                                               false, false);
}

// Sorted-descending top-16 insertion; strict-greater keeps earlier-inserted
// entries on ties (matches jax.lax.top_k smaller-index-first tie-breaking).
static __device__ __forceinline__ void topk_insert(float (&sc)[K_NB],
                                                   int (&id)[K_NB],
                                                   float d, int j) {
  if (d > sc[K_NB - 1]) {
#pragma unroll
    for (int t = K_NB - 1; t > 0; --t) {
      const bool gt  = d > sc[t];
      const bool gtp = d > sc[t - 1];          // old value (updated later)
      sc[t] = gt ? (gtp ? sc[t - 1] : d) : sc[t];
      id[t] = gt ? (gtp ? id[t - 1] : j) : id[t];
    }
    if (d > sc[0]) { sc[0] = d; id[0] = j; }
  }
}

static __device__ __forceinline__ float neg_sqdist(float4 a, float4 b) {
  const float dx = a.x - b.x, dy = a.y - b.y, dz = a.z - b.z, dw = a.w - b.w;
  return -(dx * dx + dy * dy + dz * dz + dw * dw);
}

// ---------------------------------------------------------------------------
// KNN path 1 (preferred, needs 9 MB ws): segmented scan + merge.
// Scan: thread (i, seg) scans 4096 candidates -> per-segment top-16 to ws.
// 2048 waves of independent work; j-loop unrolled x4 for load ILP. All lanes
// in a wave read the same x[j] (256 KB total -> WGP$/L2 resident, broadcast).
// ---------------------------------------------------------------------------
__global__ __launch_bounds__(64) void knn_scan_kernel(const float* __restrict__ x,
                                                      float* __restrict__ cand_sc,
                                                      int* __restrict__ cand_id) {
  const int seg = blockIdx.x >> 8;                       // 256 blocks per segment
  const int i   = ((blockIdx.x & 255) << 6) + threadIdx.x;
  const float4 xi = ((const float4*)x)[i];

  float sc[K_NB];
  int   id[K_NB];
#pragma unroll
  for (int t = 0; t < K_NB; ++t) { sc[t] = -3.402823466e38f; id[t] = 0; }

  const int j0 = seg * SEG_LEN;
#pragma unroll 4
  for (int j = j0; j < j0 + SEG_LEN; ++j) {
    const float4 xj = ((const float4*)x)[j];
    topk_insert(sc, id, neg_sqdist(xi, xj), j);
  }

  const int base = (i * NSEG + seg) * K_NB;
#pragma unroll
  for (int t = 0; t < K_NB; ++t) {
    cand_sc[base + t] = sc[t];
    cand_id[base + t] = id[t];
  }
}

// Merge: thread per row folds 4 sorted 16-lists (inserted in ascending segment
// order -> exact tie semantics preserved) into the final top-16 indices.
__global__ __launch_bounds__(64) void knn_merge_kernel(const float* __restrict__ cand_sc,
                                                       const int* __restrict__ cand_id,
                                                       int* __restrict__ idx_out) {
  const int i = (blockIdx.x << 6) + threadIdx.x;
  float sc[K_NB];
  int   id[K_NB];
#pragma unroll
  for (int t = 0; t < K_NB; ++t) { sc[t] = -3.402823466e38f; id[t] = 0; }

  const int base = i * NSEG * K_NB;
  for (int c = 0; c < NSEG * K_NB; ++c)
    topk_insert(sc, id, cand_sc[base + c], cand_id[base + c]);

#pragma unroll
  for (int t = 0; t < K_NB; ++t) idx_out[i * K_NB + t] = id[t];
}

// ---------------------------------------------------------------------------
// KNN path 2 (fallback if ws is small): single-pass scan, one thread per row.
// ---------------------------------------------------------------------------
__global__ __launch_bounds__(64) void knn_topk_kernel(const float* __restrict__ x,
                                                      int* __restrict__ idx_out) {
  const int i = (blockIdx.x << 6) + threadIdx.x;
  const float4 xi = ((const float4*)x)[i];
  float sc[K_NB];
  int   id[K_NB];
#pragma unroll
  for (int t = 0; t < K_NB; ++t) { sc[t] = -3.402823466e38f; id[t] = 0; }
#pragma unroll 4
  for (int j = 0; j < N_PTS; ++j) {
    const float4 xj = ((const float4*)x)[j];
    topk_insert(sc, id, neg_sqdist(xi, xj), j);
  }
#pragma unroll
  for (int t = 0; t < K_NB; ++t) idx_out[i * K_NB + t] = id[t];
}

// ---------------------------------------------------------------------------
// Edge features + MLP1 + MLP2 + mean, fp32 WMMA 16x16x4 (verified round 1).
// One wave per point: the 16-row tile = the point's 16 neighbors.
//   h1[16x64] = relu(edge[16x8] @ W1^T + b1)   -> 8 WMMAs
//   h2[16x64] = relu(h1 @ W2^T + b2)           -> 64 WMMAs
//   out[i]    = mean over rows (8 acc VGPRs + lane-xor-16)
// h1 round-trips LDS to convert D-layout -> A-layout. Row strides 12/68
// floats keep the lane-strided tile loads on 32 distinct banks.
// ---------------------------------------------------------------------------
__global__ __launch_bounds__(256) void edge_mlp_kernel(
    const float* __restrict__ x, const int* __restrict__ knn,
    const float* __restrict__ W1, const float* __restrict__ b1,
    const float* __restrict__ W2, const float* __restrict__ b2,
    float* __restrict__ out) {
  __shared__ float sW1[H_DIM * 12];
  __shared__ float sB1[H_DIM];
  __shared__ float sW2[H_DIM * 68];
  __shared__ float sB2[H_DIM];
  __shared__ float sH1[8 * 16 * 68];

  const int tid = threadIdx.x;
  for (int t = tid; t < H_DIM * 8; t += 256)
    sW1[(t >> 3) * 12 + (t & 7)] = W1[t];
  if (tid < H_DIM) { sB1[tid] = b1[tid]; sB2[tid] = b2[tid]; }
  for (int t = tid; t < H_DIM * H_DIM; t += 256)
    sW2[(t >> 6) * 68 + (t & 63)] = W2[t];
  __syncthreads();

  const int wave = tid >> 5;
  const int lane = tid & 31;
  const int i    = blockIdx.x * 8 + wave;
  const int m    = lane & 15;              // A/B/C col-or-row = lane%16
  const int hl   = lane >> 4;              // K-half select
  const int cb   = hl * 2;                 // lane's K pair base

  const int j = knn[i * K_NB + m];
  const float2 ctr = *(const float2*)(x + i * 4 + cb);
  const float2 nbr = *(const float2*)(x + j * 4 + cb);
  v2f A0; A0.x = nbr.x - ctr.x; A0.y = nbr.y - ctr.y;   // edge k-tile 0
  v2f A1; A1.x = ctr.x;         A1.y = ctr.y;           // edge k-tile 1

  float* myH1 = &sH1[wave * 16 * 68];

  // ---- MLP1 ----
#pragma unroll
  for (int nt = 0; nt < 4; ++nt) {
    const float bias = sB1[nt * 16 + m];
    v8f c;
#pragma unroll
    for (int r = 0; r < 8; ++r) c[r] = bias;
    const float* w1row = &sW1[(nt * 16 + m) * 12];
    v2f B0; B0.x = w1row[cb];     B0.y = w1row[cb + 1];
    v2f B1; B1.x = w1row[4 + cb]; B1.y = w1row[4 + cb + 1];
    c = wmma_f32_16x16x4(A0, B0, c);
    c = wmma_f32_16x16x4(A1, B1, c);
#pragma unroll
    for (int r = 0; r < 8; ++r) {
      const float v = c[r] > 0.0f ? c[r] : 0.0f;
      myH1[(r + 8 * hl) * 68 + nt * 16 + m] = v;
    }
  }
  __syncthreads();

  // ---- MLP2 + mean ----
#pragma unroll
  for (int g = 0; g < 4; ++g) {
    const float bias2 = sB2[g * 16 + m];
    v8f c;
#pragma unroll
    for (int r = 0; r < 8; ++r) c[r] = bias2;
    const float* w2row = &sW2[(g * 16 + m) * 68];
    const float* h1row = &myH1[m * 68];
#pragma unroll
    for (int kt = 0; kt < 16; ++kt) {
      const int hh = kt * 4 + cb;
      v2f a; a.x = h1row[hh]; a.y = h1row[hh + 1];
      v2f b; b.x = w2row[hh]; b.y = w2row[hh + 1];
      c = wmma_f32_16x16x4(a, b, c);
    }
    float s = 0.0f;
#pragma unroll
    for (int r = 0; r < 8; ++r) s += (c[r] > 0.0f ? c[r] : 0.0f);
    s += __shfl_xor(s, 16, 32);
    s *= (1.0f / 16.0f);
    if (lane < 16) out[i * H_DIM + g * 16 + lane] = s;
  }
}

extern "C" void kernel_launch(void* const* d_in, const int* in_sizes, int n_in,
                              void* d_out, int out_size, void* d_ws, size_t ws_size,
                              hipStream_t stream) {
  (void)in_sizes; (void)n_in; (void)out_size;
  const float* x  = (const float*)d_in[0];
  const float* W1 = (const float*)d_in[1];
  const float* b1 = (const float*)d_in[2];
  const float* W2 = (const float*)d_in[3];
  const float* b2 = (const float*)d_in[4];
  float* out = (float*)d_out;

  // ws layout: [0,1MB) final idx | [1MB,5MB) cand scores | [5MB,9MB) cand ids
  char* ws = (char*)d_ws;
  int*   knn     = (int*)ws;
  float* cand_sc = (float*)(ws + (size_t)N_PTS * K_NB * 4);
  int*   cand_id = (int*)(ws + (size_t)N_PTS * K_NB * 4 +
                          (size_t)N_PTS * NSEG * K_NB * 4);
  const size_t ws_needed = (size_t)N_PTS * K_NB * 4 +
                           2 * (size_t)N_PTS * NSEG * K_NB * 4;

  if (ws_size >= ws_needed) {
    knn_scan_kernel<<<(N_PTS / 64) * NSEG, 64, 0, stream>>>(x, cand_sc, cand_id);
    knn_merge_kernel<<<N_PTS / 64, 64, 0, stream>>>(cand_sc, cand_id, knn);
  } else {
    knn_topk_kernel<<<N_PTS / 64, 64, 0, stream>>>(x, knn);
  }
  edge_mlp_kernel<<<N_PTS / 8, 256, 0, stream>>>(x, knn, W1, b1, W2, b2, out);
}